// CrissCrossAttention3D_6047313953130
// MI455X (gfx1250) — compile-verified
//
#include <hip/hip_runtime.h>
#include <math.h>

// Problem constants (B=1)
#define HD 64
#define WD 64
#define DD 64
#define CQN 8
#define CVN 64
#define CHS (HD*WD*DD)   // channel stride = 262144 (same for Q/K/V/out)
#define SH  (WD*DD)      // 4096  (h stride)
#define SW  (DD)         // 64    (w stride)
#define DT  4            // d-slices per block in the H/W passes

typedef float v2f __attribute__((ext_vector_type(2)));
typedef float v8f __attribute__((ext_vector_type(8)));

// ---------------------------------------------------------------------------
// Pass 0: per-point softmax stats (max, sum of exp) over all 192 energies.
// One block per (h,w), one thread per d. K fits in L2 (8.4MB) so the direct
// global reads (coalesced in d) are cheap.
// ---------------------------------------------------------------------------
__global__ __launch_bounds__(64) void cca3d_stats(const float* __restrict__ Q,
                                                  const float* __restrict__ K,
                                                  float* __restrict__ smax,
                                                  float* __restrict__ ssum) {
    const int h = blockIdx.x / WD;
    const int w = blockIdx.x % WD;
    const int d = threadIdx.x;

    __shared__ float Kd[CQN][DD];   // K[c,h,w,z] : shared D-line
    for (int i = threadIdx.x; i < CQN * DD; i += 64) {
        int c = i >> 6, z = i & 63;
        Kd[c][z] = K[c * CHS + h * SH + w * SW + z];
    }
    __syncthreads();

    const int base = h * SH + w * SW + d;
    float q[CQN];
#pragma unroll
    for (int c = 0; c < CQN; ++c) q[c] = Q[c * CHS + base];

    float m = -INFINITY, s = 0.0f;
    // H direction: e = <q, K[:,x,w,d]>, diag x==h masked
    for (int x = 0; x < HD; ++x) {
        if (x == h) continue;
        float e = 0.0f;
#pragma unroll
        for (int c = 0; c < CQN; ++c) e = fmaf(q[c], K[c * CHS + x * SH + w * SW + d], e);
        float nm = fmaxf(m, e);
        s = s * __expf(m - nm) + __expf(e - nm);
        m = nm;
    }
    // W direction: unmasked (matches reference)
    for (int y = 0; y < WD; ++y) {
        float e = 0.0f;
#pragma unroll
        for (int c = 0; c < CQN; ++c) e = fmaf(q[c], K[c * CHS + h * SH + y * SW + d], e);
        float nm = fmaxf(m, e);
        s = s * __expf(m - nm) + __expf(e - nm);
        m = nm;
    }
    // D direction: diag z==d masked
    for (int z = 0; z < DD; ++z) {
        if (z == d) continue;
        float e = 0.0f;
#pragma unroll
        for (int c = 0; c < CQN; ++c) e = fmaf(q[c], Kd[c][z], e);
        float nm = fmaxf(m, e);
        s = s * __expf(m - nm) + __expf(e - nm);
        m = nm;
    }
    smax[base] = m;
    ssum[base] = s;
}

// ---------------------------------------------------------------------------
// Pass D: block per (h,w). Builds att_D(64x64) in LDS, then
// out[c,d] = sum_z V[c,h,w,z] * att[d,z] as a 64x64x64 fp32 WMMA GEMM.
// Writes (initializes) out, coalesced along d.
// ---------------------------------------------------------------------------
__global__ __launch_bounds__(256) void cca3d_passD(const float* __restrict__ Q,
                                                   const float* __restrict__ K,
                                                   const float* __restrict__ V,
                                                   const float* __restrict__ smax,
                                                   const float* __restrict__ ssum,
                                                   float* __restrict__ out) {
    __shared__ float Vt[CVN][DD + 1];   // V[c,h,w,z]
    __shared__ float At[DD][DD + 1];    // att[d][z]
    __shared__ float Kl[CQN][DD];
    __shared__ float Ql[CQN][DD];
    __shared__ float sM[DD], sR[DD];

    const int h = blockIdx.x / WD;
    const int w = blockIdx.x % WD;
    const int base = h * SH + w * SW;
    const int tid = threadIdx.x;

    for (int i = tid; i < CVN * DD; i += 256) {
        int c = i >> 6, z = i & 63;
        Vt[c][z] = V[c * CHS + base + z];
    }
    for (int i = tid; i < CQN * DD; i += 256) {
        int c = i >> 6, z = i & 63;
        Kl[c][z] = K[c * CHS + base + z];
        Ql[c][z] = Q[c * CHS + base + z];
    }
    if (tid < DD) { sM[tid] = smax[base + tid]; sR[tid] = 1.0f / ssum[base + tid]; }
    __syncthreads();

    for (int i = tid; i < DD * DD; i += 256) {
        int dd = i >> 6, z = i & 63;
        float e = 0.0f;
#pragma unroll
        for (int c = 0; c < CQN; ++c) e = fmaf(Ql[c][dd], Kl[c][z], e);
        At[dd][z] = (z == dd) ? 0.0f : __expf(e - sM[dd]) * sR[dd];
    }
    __syncthreads();

    const int wave = tid >> 5, lane = tid & 31;
    const int hi = lane >> 4, ln = lane & 15;
    for (int tile = wave; tile < 16; tile += 8) {
        const int mi = tile >> 2, ni = tile & 3;
        v8f acc = {};
#pragma unroll
        for (int kk = 0; kk < 16; ++kk) {
            const int k0 = kk * 4 + hi * 2;          // k = vgpr + 2*(lane>=16)
            v2f a = { Vt[mi * 16 + ln][k0], Vt[mi * 16 + ln][k0 + 1] };
            v2f b = { At[ni * 16 + ln][k0], At[ni * 16 + ln][k0 + 1] };
            acc = __builtin_amdgcn_wmma_f32_16x16x4_f32(false, a, false, b,
                                                        (short)0, acc, false, false);
        }
#pragma unroll
        for (int r = 0; r < 8; ++r) {
            const int c = mi * 16 + r + 8 * hi;      // C/D row = vgpr + 8*(lane>=16)
            const int dc = ni * 16 + ln;
            out[c * CHS + base + dc] = acc[r];
        }
    }
}

// ---------------------------------------------------------------------------
// Passes H and W (unified). Block covers fixed coord f and DT=4 d-slices.
//   addr(c, t_or_p, j) = c*CHS + t*s + f*fs + d0 + j
// H-pass: s=SH (t=x, p=h), fs=SW (f=w), mask x==h.
// W-pass: s=SW (t=y, p=w), fs=SH (f=h), no mask.
// out is accumulated with aligned float4 read-modify-write.
// ---------------------------------------------------------------------------
__global__ __launch_bounds__(256) void cca3d_passHW(const float* __restrict__ Q,
                                                    const float* __restrict__ K,
                                                    const float* __restrict__ V,
                                                    const float* __restrict__ smax,
                                                    const float* __restrict__ ssum,
                                                    float* __restrict__ out,
                                                    int s, int fs, int applyMask) {
    __shared__ float Vt[DT][CVN][65];
    __shared__ float At[DT][64][65];
    __shared__ float Kl[DT][CQN][64];
    __shared__ float Ql[DT][CQN][64];
    __shared__ float sM[DT][64], sR[DT][64];

    const int f  = blockIdx.x >> 4;           // DD/DT = 16
    const int d0 = (blockIdx.x & 15) * DT;
    const int tid = threadIdx.x;
    const int fbase = f * fs + d0;

    for (int i = tid; i < CVN * 64; i += 256) {
        int c = i >> 6, t = i & 63;
        float4 v = *(const float4*)(V + c * CHS + t * s + fbase);
        Vt[0][c][t] = v.x; Vt[1][c][t] = v.y; Vt[2][c][t] = v.z; Vt[3][c][t] = v.w;
    }
    for (int i = tid; i < CQN * 64; i += 256) {
        int c = i >> 6, t = i & 63;
        float4 kv = *(const float4*)(K + c * CHS + t * s + fbase);
        float4 qv = *(const float4*)(Q + c * CHS + t * s + fbase);
        Kl[0][c][t] = kv.x; Kl[1][c][t] = kv.y; Kl[2][c][t] = kv.z; Kl[3][c][t] = kv.w;
        Ql[0][c][t] = qv.x; Ql[1][c][t] = qv.y; Ql[2][c][t] = qv.z; Ql[3][c][t] = qv.w;
    }
    if (tid < 64) {
        float4 mv = *(const float4*)(smax + tid * s + fbase);
        float4 sv = *(const float4*)(ssum + tid * s + fbase);
        sM[0][tid] = mv.x; sM[1][tid] = mv.y; sM[2][tid] = mv.z; sM[3][tid] = mv.w;
        sR[0][tid] = 1.0f / sv.x; sR[1][tid] = 1.0f / sv.y;
        sR[2][tid] = 1.0f / sv.z; sR[3][tid] = 1.0f / sv.w;
    }
    __syncthreads();

    for (int i = tid; i < DT * 64 * 64; i += 256) {
        int j = i >> 12, p = (i >> 6) & 63, t = i & 63;
        float e = 0.0f;
#pragma unroll
        for (int c = 0; c < CQN; ++c) e = fmaf(Ql[j][c][p], Kl[j][c][t], e);
        float a = __expf(e - sM[j][p]) * sR[j][p];
        if (applyMask && t == p) a = 0.0f;
        At[j][p][t] = a;
    }
    __syncthreads();

    const int wave = tid >> 5, lane = tid & 31;
    const int hi = lane >> 4, ln = lane & 15;
    for (int tile = wave; tile < 16; tile += 8) {
        const int mi = tile >> 2, ni = tile & 3;
        v8f acc[DT] = {};
#pragma unroll
        for (int kk = 0; kk < 16; ++kk) {
            const int k0 = kk * 4 + hi * 2;
#pragma unroll
            for (int j = 0; j < DT; ++j) {
                v2f a = { Vt[j][mi * 16 + ln][k0], Vt[j][mi * 16 + ln][k0 + 1] };
                v2f b = { At[j][ni * 16 + ln][k0], At[j][ni * 16 + ln][k0 + 1] };
                acc[j] = __builtin_amdgcn_wmma_f32_16x16x4_f32(false, a, false, b,
                                                               (short)0, acc[j], false, false);
            }
        }
#pragma unroll
        for (int r = 0; r < 8; ++r) {
            const int c = mi * 16 + r + 8 * hi;
            const int p = ni * 16 + ln;
            float4* po = (float4*)(out + c * CHS + p * s + fbase);
            float4 o = *po;
            o.x += acc[0][r]; o.y += acc[1][r]; o.z += acc[2][r]; o.w += acc[3][r];
            *po = o;
        }
    }
}

// ---------------------------------------------------------------------------
extern "C" void kernel_launch(void* const* d_in, const int* in_sizes, int n_in,
                              void* d_out, int out_size, void* d_ws, size_t ws_size,
                              hipStream_t stream) {
    (void)in_sizes; (void)n_in; (void)out_size; (void)ws_size;
    const float* Q = (const float*)d_in[0];
    const float* K = (const float*)d_in[1];
    const float* V = (const float*)d_in[2];
    float* out = (float*)d_out;
    float* smax = (float*)d_ws;          // 262144 floats
    float* ssum = smax + CHS;            // 262144 floats  (needs 2MB of ws)

    cca3d_stats <<<HD * WD, 64,  0, stream>>>(Q, K, smax, ssum);
    cca3d_passD <<<HD * WD, 256, 0, stream>>>(Q, K, V, smax, ssum, out);            // init out
    cca3d_passHW<<<WD * (DD / DT), 256, 0, stream>>>(Q, K, V, smax, ssum, out,
                                                     SH, SW, 1);                    // H (masked)
    cca3d_passHW<<<HD * (DD / DT), 256, 0, stream>>>(Q, K, V, smax, ssum, out,
                                                     SW, SH, 0);                    // W
}